// SimplifiedGCN_5574867550498
// MI455X (gfx1250) — compile-verified
//
#include <hip/hip_runtime.h>

typedef __attribute__((ext_vector_type(2))) float v2f;
typedef __attribute__((ext_vector_type(8))) float v8f;

#define F_IN 512
#define HID  16
#define NCLS 64

// ---------------- degree / normalization ----------------

__global__ void k_deg_init(float* deg, int n) {
    int i = blockIdx.x * blockDim.x + threadIdx.x;
    if (i < n) deg[i] = 1.0f;   // self-loop contribution
}

__global__ void k_deg_edges(const long long* __restrict__ dst, float* deg, int E) {
    int e = blockIdx.x * blockDim.x + threadIdx.x;
    if (e < E) atomicAdd(&deg[(int)dst[e]], 1.0f);
}

__global__ void k_dinv(float* deg, int n) {
    int i = blockIdx.x * blockDim.x + threadIdx.x;
    if (i < n) deg[i] = rsqrtf(deg[i]);   // deg >= 1 always (self-loops)
}

// ---------------- GEMM1: h0 = x @ W1 via V_WMMA_F32_16X16X4_F32 ----------------
// One wave per 16-row M-tile, N = HID = 16 (one tile wide), K-loop of 512/4.

__global__ void k_gemm1_wmma(const float* __restrict__ x,
                             const float* __restrict__ W1,
                             float* __restrict__ h0, int ntiles) {
    int wave = threadIdx.x >> 5;
    int lane = threadIdx.x & 31;
    int tile = blockIdx.x * 4 + wave;
    if (tile >= ntiles) return;              // wave-uniform: EXEC stays all-1s

    int khalf = (lane >> 4) << 1;            // lanes 0-15 -> K {0,1}; 16-31 -> K {2,3}
    int nidx  = lane & 15;
    const float* xrow = x + (size_t)(tile * 16 + nidx) * F_IN;

    v8f c = {};
    for (int k = 0; k < F_IN; k += 4) {
        v2f a, b;
        a.x = xrow[k + khalf];
        a.y = xrow[k + khalf + 1];
        b.x = W1[(k + khalf)     * HID + nidx];
        b.y = W1[(k + khalf + 1) * HID + nidx];
        c = __builtin_amdgcn_wmma_f32_16x16x4_f32(
                false, a, false, b, (short)0, c, false, false);
    }
    // C layout: VGPR j -> row j (lanes 0-15) / row j+8 (lanes 16-31), col = lane&15
    int m0 = tile * 16 + ((lane >> 4) << 3);
#pragma unroll
    for (int j = 0; j < 8; ++j)
        h0[(size_t)(m0 + j) * HID + nidx] = c[j];
}

// ---------------- GEMM2: h2 = h1 @ W2 (16 -> 64) ----------------
// One block per M-tile; wave w handles N-tile w (n0 = 16w); K = 16 -> 4 WMMAs.

__global__ void k_gemm2_wmma(const float* __restrict__ h1,
                             const float* __restrict__ W2,
                             float* __restrict__ h2) {
    int wave = threadIdx.x >> 5;             // N-tile 0..3
    int lane = threadIdx.x & 31;
    int mtile = blockIdx.x;
    int n0 = wave << 4;
    int khalf = (lane >> 4) << 1;
    int nidx  = lane & 15;
    const float* arow = h1 + (size_t)(mtile * 16 + nidx) * HID;

    v8f c = {};
#pragma unroll
    for (int k = 0; k < HID; k += 4) {
        v2f a, b;
        a.x = arow[k + khalf];
        a.y = arow[k + khalf + 1];
        b.x = W2[(k + khalf)     * NCLS + n0 + nidx];
        b.y = W2[(k + khalf + 1) * NCLS + n0 + nidx];
        c = __builtin_amdgcn_wmma_f32_16x16x4_f32(
                false, a, false, b, (short)0, c, false, false);
    }
    int m0 = mtile * 16 + ((lane >> 4) << 3);
#pragma unroll
    for (int j = 0; j < 8; ++j)
        h2[(size_t)(m0 + j) * NCLS + n0 + nidx] = c[j];
}

// ---------------- propagation ----------------
// init: hout = bias + dinv[i]^2 * hin   (self-loop + bias, fully overwrites hout)

template <int F>
__global__ void k_self_init(const float* __restrict__ hin,
                            const float* __restrict__ dinv,
                            const float* __restrict__ bias,
                            float* __restrict__ hout, int n) {
    long long idx = (long long)blockIdx.x * blockDim.x + threadIdx.x;
    if (idx >= (long long)n * F) return;
    int i = (int)(idx / F);
    int f = (int)(idx % F);
    float w = dinv[i] * dinv[i];
    hout[idx] = bias[f] + w * hin[idx];
}

// edges: hout[dst] += dinv[s]*dinv[d] * hin[src]; F consecutive lanes per edge.

template <int F>
__global__ void k_scatter_edges(const long long* __restrict__ src,
                                const long long* __restrict__ dst,
                                const float* __restrict__ hin,
                                const float* __restrict__ dinv,
                                float* __restrict__ hout, int E) {
    long long idx = (long long)blockIdx.x * blockDim.x + threadIdx.x;
    if (idx >= (long long)E * F) return;
    int e = (int)(idx / F);
    int f = (int)(idx % F);
    int s = (int)src[e];
    int d = (int)dst[e];
    float w = dinv[s] * dinv[d];
    atomicAdd(&hout[(size_t)d * F + f], w * hin[(size_t)s * F + f]);
}

__global__ void k_relu(float* h, long long total) {
    long long idx = (long long)blockIdx.x * blockDim.x + threadIdx.x;
    if (idx < total) h[idx] = fmaxf(h[idx], 0.0f);
}

// ---------------- launch ----------------

extern "C" void kernel_launch(void* const* d_in, const int* in_sizes, int n_in,
                              void* d_out, int out_size, void* d_ws, size_t ws_size,
                              hipStream_t stream) {
    const float*      x  = (const float*)d_in[0];
    const long long*  ei = (const long long*)d_in[1];   // int64 [2, E]
    const float*      W1 = (const float*)d_in[2];
    const float*      b1 = (const float*)d_in[3];
    const float*      W2 = (const float*)d_in[4];
    const float*      b2 = (const float*)d_in[5];
    float*            out = (float*)d_out;

    const int N = in_sizes[0] / F_IN;       // 100000
    const int E = in_sizes[1] / 2;          // 3200000
    const long long* src = ei;
    const long long* dst = ei + E;

    // workspace layout (floats): dinv | h0 (N*16) | h1 (N*16) | h2 (N*64)
    float* ws   = (float*)d_ws;
    float* dinv = ws;
    float* h0   = ws + (((size_t)N + 255) & ~(size_t)255);
    float* h1   = h0 + (size_t)N * HID;
    float* h2   = h1 + (size_t)N * HID;

    const int T = 256;

    // degrees -> dinv
    k_deg_init <<<(N + T - 1) / T, T, 0, stream>>>(dinv, N);
    k_deg_edges<<<(E + T - 1) / T, T, 0, stream>>>(dst, dinv, E);
    k_dinv     <<<(N + T - 1) / T, T, 0, stream>>>(dinv, N);

    // layer 1
    int mt = N / 16;                         // 6250 tiles
    k_gemm1_wmma<<<(mt + 3) / 4, 128, 0, stream>>>(x, W1, h0, mt);

    long long t1  = (long long)N * HID;
    long long te1 = (long long)E * HID;
    k_self_init<HID>    <<<(int)((t1  + T - 1) / T), T, 0, stream>>>(h0, dinv, b1, h1, N);
    k_scatter_edges<HID><<<(int)((te1 + T - 1) / T), T, 0, stream>>>(src, dst, h0, dinv, h1, E);
    k_relu              <<<(int)((t1  + T - 1) / T), T, 0, stream>>>(h1, t1);

    // layer 2
    k_gemm2_wmma<<<mt, 128, 0, stream>>>(h1, W2, h2);

    long long t2  = (long long)N * NCLS;
    long long te2 = (long long)E * NCLS;
    k_self_init<NCLS>    <<<(int)((t2  + T - 1) / T), T, 0, stream>>>(h2, dinv, b2, out, N);
    k_scatter_edges<NCLS><<<(int)((te2 + T - 1) / T), T, 0, stream>>>(src, dst, h2, dinv, out, E);
}